// crossbar_34514357191144
// MI455X (gfx1250) — compile-verified
//
#include <hip/hip_runtime.h>
#include <hip/hip_bf16.h>

typedef __attribute__((ext_vector_type(2))) float v2f;
typedef __attribute__((ext_vector_type(8))) float v8f;

// ---- problem constants (match reference) ----
#define OUT_CH   256
#define IN_CH    128
#define KK       3
#define ROW_LEN  (IN_CH * KK * KK)   // 1152 = 18*64, no padding needed
#define NBATCH   16
#define IMG      32
#define LTOT     (IMG * IMG)         // 1024
#define CRXB     64
#define CRXB_ROW 18
#define CRXB_COL 4
#define HALF_LVL 127
#define GMAXC    (1.0f / 3000.0f)
#define GMINC    (1.0f / 3000000.0f)
#define DELTA_G  ((GMAXC - GMINC) / (float)HALF_LVL)

#define WSTRIDE  66     // 64 + 2 pad: A-fragment b64 loads spread across banks
#define LTILE    128    // l-columns per block
#define NLTILE   (LTOT / LTILE)      // 8
// U stored pair-interleaved: [kpair][col][2], kpair = 0..31, col = 0..127
// stride 288 floats: 288 % 64 == 32 -> the two row-halves of a wave use
// disjoint 32-bank halves -> conflict-free b64 B-fragment loads.
#define UPSTRIDE 288

// ---------------- pass 0/1: |W| max reduction ----------------
__global__ void init_ws_kernel(unsigned* ws) {
    if (blockIdx.x == 0 && threadIdx.x == 0) ws[0] = 0u;
}

__global__ __launch_bounds__(256) void absmax_kernel(const float* __restrict__ w,
                                                     int n, unsigned* __restrict__ ws) {
    __shared__ float smax[256];
    float m = 0.0f;
    for (int i = blockIdx.x * blockDim.x + threadIdx.x; i < n; i += gridDim.x * blockDim.x)
        m = fmaxf(m, fabsf(w[i]));
    smax[threadIdx.x] = m;
    __syncthreads();
    for (int s = 128; s > 0; s >>= 1) {
        if (threadIdx.x < s)
            smax[threadIdx.x] = fmaxf(smax[threadIdx.x], smax[threadIdx.x + s]);
        __syncthreads();
    }
    if (threadIdx.x == 0)
        atomicMax(ws, __float_as_uint(smax[0]));   // values >= 0: uint order == float order
}

// im2col element: U[b, g, l] with g = ch*9 + ky*3 + kx, l = y*32 + x
__device__ __forceinline__ float im2col_elem(const float* __restrict__ x,
                                             int b, int g, int l) {
    int ch  = g / 9;
    int rem = g - ch * 9;
    int ky  = rem / 3;
    int kx  = rem - ky * 3;
    int y   = (l >> 5) + ky - 1;
    int xx  = (l & 31) + kx - 1;
    if ((unsigned)y < (unsigned)IMG && (unsigned)xx < (unsigned)IMG)
        return x[(((size_t)b * IN_CH + ch) * IMG + y) * IMG + xx];
    return 0.0f;
}

// ---------------- pass 2: fused im2col + quantize + block GEMM ----------------
// grid.x = NBATCH * CRXB_ROW * NLTILE = 2304, block = 256 (8 waves)
__global__ __launch_bounds__(256) void crossbar_gemm_kernel(const float* __restrict__ x,
                                                            const float* __restrict__ weight,
                                                            float* __restrict__ out,
                                                            const unsigned* __restrict__ ws) {
    __shared__ __align__(16) float ldsW[64 * WSTRIDE];     // 16.9 KB
    __shared__ __align__(16) float ldsU[32 * UPSTRIDE];    // 36.9 KB, pair-interleaved

    const int blk   = blockIdx.x;
    const int ltile = blk & (NLTILE - 1);
    const int r     = (blk >> 3) % CRXB_ROW;
    const int b     = blk / (NLTILE * CRXB_ROW);

    const float maxab   = __uint_as_float(ws[0]);
    const float delta_w = maxab / (float)HALF_LVL;

    const int tid    = threadIdx.x;
    const int l0base = ltile * LTILE;

    // ---- stage im2col tile into LDS, row-pair interleaved ----
    // pair p (= rows 2p, 2p+1), col lx:  ldsU[p*UPSTRIDE + 2*lx + {0,1}]
    #pragma unroll
    for (int it = 0; it < (32 * LTILE) / 256; ++it) {
        int e  = tid + it * 256;
        int p  = e >> 7;           // 0..31
        int lx = e & (LTILE - 1);  // 0..127
        int g0 = r * CRXB + 2 * p;
        int l  = l0base + lx;
        v2f u;
        u.x = im2col_elem(x, b, g0,     l);
        u.y = im2col_elem(x, b, g0 + 1, l);
        *(v2f*)&ldsU[p * UPSTRIDE + 2 * lx] = u;   // single ds_store_b64
    }

    const int wv    = tid >> 5;
    const int lane  = tid & 31;
    const int lhalf = lane >> 4;     // 0: lanes 0-15, 1: lanes 16-31
    const int lmod  = lane & 15;
    const int i0    = (wv >> 1) * 16;   // output row group of this wave
    const int lw    = (wv & 1) * 64;    // output col half of this wave

    for (int c = 0; c < CRXB_COL; ++c) {
        __syncthreads();
        // ---- stage quantized, DELTA_G-scaled weight tile W[c, r] (64x64) ----
        #pragma unroll
        for (int it = 0; it < (64 * 64) / 256; ++it) {
            int e = tid + it * 256;
            int i = e >> 6;
            int j = e & 63;
            float wv_ = weight[(size_t)(c * CRXB + i) * ROW_LEN + r * CRXB + j];
            ldsW[i * WSTRIDE + j] = rintf(wv_ / delta_w) * DELTA_G;
        }
        __syncthreads();

        v8f acc0 = {}, acc1 = {}, acc2 = {}, acc3 = {};
        #pragma unroll
        for (int k0 = 0; k0 < CRXB; k0 += 4) {
            // A fragment (16x4 f32): lane<16 -> K=k0,k0+1 ; lane>=16 -> K=k0+2,k0+3
            const float* ap = &ldsW[(i0 + lmod) * WSTRIDE + k0 + 2 * lhalf];
            v2f a = *(const v2f*)ap;
            // B fragments (4x16 f32): one aligned b64 per tile from the
            // pair-interleaved U: rows {k0+2*lhalf, k0+2*lhalf+1} at col.
            const float* bp = &ldsU[((k0 >> 1) + lhalf) * UPSTRIDE + 2 * (lw + lmod)];
            v2f b0 = *(const v2f*)(bp);
            v2f b1 = *(const v2f*)(bp + 32);
            v2f b2 = *(const v2f*)(bp + 64);
            v2f b3 = *(const v2f*)(bp + 96);
            acc0 = __builtin_amdgcn_wmma_f32_16x16x4_f32(false, a, false, b0, (short)0, acc0, false, false);
            acc1 = __builtin_amdgcn_wmma_f32_16x16x4_f32(false, a, false, b1, (short)0, acc1, false, false);
            acc2 = __builtin_amdgcn_wmma_f32_16x16x4_f32(false, a, false, b2, (short)0, acc2, false, false);
            acc3 = __builtin_amdgcn_wmma_f32_16x16x4_f32(false, a, false, b3, (short)0, acc3, false, false);
        }

        // ---- store: C/D layout lane(vgpr v) -> M = 8*lhalf + v, N = lmod ----
        size_t obase = ((((size_t)b * CRXB_COL + c) * CRXB_ROW + r) * CRXB + i0) * (size_t)LTOT
                       + l0base + lw;
        #pragma unroll
        for (int v = 0; v < 8; ++v) {
            size_t rowoff = (size_t)(8 * lhalf + v) * LTOT + lmod;
            out[obase + rowoff]      = acc0[v];
            out[obase + rowoff + 16] = acc1[v];
            out[obase + rowoff + 32] = acc2[v];
            out[obase + rowoff + 48] = acc3[v];
        }
        __syncthreads();   // protect ldsW before next c overwrites it
    }
}

extern "C" void kernel_launch(void* const* d_in, const int* in_sizes, int n_in,
                              void* d_out, int out_size, void* d_ws, size_t ws_size,
                              hipStream_t stream) {
    const float* x      = (const float*)d_in[0];   // [16,128,32,32] f32
    const float* weight = (const float*)d_in[1];   // [256,128,3,3]  f32
    float* out          = (float*)d_out;           // [16,4,18,64,1024] f32
    unsigned* ws        = (unsigned*)d_ws;

    const int n_w = OUT_CH * IN_CH * KK * KK;      // 294912

    init_ws_kernel<<<1, 1, 0, stream>>>(ws);
    absmax_kernel<<<256, 256, 0, stream>>>(weight, n_w, ws);
    crossbar_gemm_kernel<<<NBATCH * CRXB_ROW * NLTILE, 256, 0, stream>>>(x, weight, out, ws);
}